// RoutingLayer_84275848282528
// MI455X (gfx1250) — compile-verified
//
#include <hip/hip_runtime.h>

typedef __attribute__((ext_vector_type(2))) float v2f;
typedef __attribute__((ext_vector_type(8))) float v8f;

#define B_TOT   2048
#define I_CAP   64
#define O_CAP   32
#define D_DIM   16          // DI == DO == 16
#define BT      16          // b-tile (WMMA M)
#define NBT     (B_TOT / BT)   // 128
#define PR_PITCH 68         // padded i-pitch in floats (bank-conflict avoidance)
#define PR_BYTES (16 * 16 * PR_PITCH * 4)   // 69632 bytes LDS

// ---------------------------------------------------------------------------
// Pack x (B, DI, I) f32 into WMMA A-fragment order:
//   A tile per (btile, i): A[m=b_local, k=e], 16x16 split into 4 K-chunks of 4.
//   Lane layout (16x4 f32 A): lane L: m = L&15; vgpr j holds K = 2*(L>>4) + j.
//   xp[btile][i][kk][lane] = float2{ xt[b, i, e0], xt[b, i, e0+1] },
//   e0 = kk*4 + 2*(lane>>4),  xt[b,i,e] = x[b*1024 + e*64 + i].
// ---------------------------------------------------------------------------
__global__ __launch_bounds__(256) void pack_x_kernel(const float* __restrict__ x,
                                                     v2f* __restrict__ xp)
{
    int f    = blockIdx.x * 256 + threadIdx.x;   // < 128*64*4*32 = 1048576
    int lane = f & 31;
    int kk   = (f >> 5) & 3;
    int i    = (f >> 7) & 63;
    int bt   = f >> 13;
    int m    = lane & 15;
    int e0   = kk * 4 + ((lane >> 4) << 1);
    const float* s = x + (size_t)(bt * BT + m) * 1024 + i;
    v2f v;
    v.x = s[(size_t)e0 * 64];
    v.y = s[(size_t)(e0 + 1) * 64];
    xp[f] = v;
}

// ---------------------------------------------------------------------------
// Pack W = rw + noise, (O, I, DO, DI) f32, into WMMA B-fragment order:
//   B tile per (o, i): Bmat[k=e, n=d] = W[o,i,d,e].
//   Lane layout (4x16 f32 B): lane L: n = L&15; vgpr j holds K = 2*(L>>4) + j.
// ---------------------------------------------------------------------------
__global__ __launch_bounds__(256) void pack_w_kernel(const float* __restrict__ rw,
                                                     const float* __restrict__ nz,
                                                     v2f* __restrict__ wf)
{
    int f    = blockIdx.x * 256 + threadIdx.x;   // < 32*64*4*32 = 262144
    int lane = f & 31;
    int kk   = (f >> 5) & 3;
    int i    = (f >> 7) & 63;
    int o    = f >> 13;
    int d    = lane & 15;
    int e0   = kk * 4 + ((lane >> 4) << 1);
    size_t idx = ((size_t)(o * I_CAP + i) * 16 + d) * 16 + e0;
    v2f v;
    v.x = rw[idx]     + nz[idx];
    v.y = rw[idx + 1] + nz[idx + 1];
    wf[f] = v;
}

// ---------------------------------------------------------------------------
// Fused priors-GEMM (fp32 WMMA) + 3-iteration dynamic routing.
// Block = 256 threads (8 wave32), grid = (128 b-tiles, 32 o).
// Wave w computes priors for i in [8w, 8w+8) via chained v_wmma_f32_16x16x4_f32,
// scatters D into LDS pr[b][d][i] (pitch-68), then each thread owns one (b,d)
// pair: softmax over i is a serial register loop, squash reduces over d via
// 16-lane shfl_xor tree.
// ---------------------------------------------------------------------------
__global__ __launch_bounds__(256) void routing_fused_kernel(const v2f* __restrict__ xp,
                                                            const v2f* __restrict__ wf,
                                                            float* __restrict__ out)
{
    extern __shared__ float pr[];   // [16 b][16 d][PR_PITCH]

    const int bt   = blockIdx.x;
    const int o    = blockIdx.y;
    const int tid  = threadIdx.x;
    const int lane = tid & 31;
    const int wv   = tid >> 5;
    const int dn   = lane & 15;           // WMMA N index (= d)
    const int hi8  = (lane >> 4) << 3;    // +8 rows for upper half-wave

    // ---- compute phase: priors[b, i, d] for this (o, b-tile) ----
    #pragma unroll
    for (int ii = 0; ii < 8; ++ii) {
        const int i = wv * 8 + ii;
        const v2f* pa = xp + (((size_t)bt * I_CAP + i) * 4) * 32 + lane;
        const v2f* pb = wf + (((size_t)o  * I_CAP + i) * 4) * 32 + lane;
        v8f acc = {};
        #pragma unroll
        for (int kk = 0; kk < 4; ++kk) {
            v2f a = pa[kk * 32];
            v2f b = pb[kk * 32];
            // D = A(16x4) x B(4x16) + C : accumulate over e in chunks of 4
            acc = __builtin_amdgcn_wmma_f32_16x16x4_f32(
                false, a, false, b, (short)0, acc, false, false);
        }
        #pragma unroll
        for (int v = 0; v < 8; ++v) {
            const int bl = v + hi8;       // D-matrix row: M = v (+8 for lanes 16-31)
            pr[(bl * 16 + dn) * PR_PITCH + i] = acc[v];
        }
    }
    __syncthreads();

    // ---- routing phase: thread <-> (b, d) pair ----
    const int tb = tid >> 4;
    const int td = tid & 15;

    float prv[64];
    const float4* row = reinterpret_cast<const float4*>(&pr[(tb * 16 + td) * PR_PITCH]);
    #pragma unroll
    for (int q = 0; q < 16; ++q) {
        float4 fq = row[q];
        prv[4 * q + 0] = fq.x;  prv[4 * q + 1] = fq.y;
        prv[4 * q + 2] = fq.z;  prv[4 * q + 3] = fq.w;
    }

    float lg[64];
    #pragma unroll
    for (int i = 0; i < 64; ++i) lg[i] = 0.0f;

    float outd = 0.0f;
    #pragma unroll
    for (int it = 0; it < 3; ++it) {
        // softmax over i (serial per thread), fused with weighted prior sum
        float mx = lg[0];
        #pragma unroll
        for (int i = 1; i < 64; ++i) mx = fmaxf(mx, lg[i]);
        float se = 0.0f, s = 0.0f;
        #pragma unroll
        for (int i = 0; i < 64; ++i) {
            float e = __expf(lg[i] - mx);
            se += e;
            s = fmaf(e, prv[i], s);
        }
        s /= se;

        // squash: sn = sum over d (16 lanes) of s^2
        float sn = s * s;
        sn += __shfl_xor(sn, 1, 32);
        sn += __shfl_xor(sn, 2, 32);
        sn += __shfl_xor(sn, 4, 32);
        sn += __shfl_xor(sn, 8, 32);
        outd = (sn / (1.0f + sn)) * rsqrtf(sn) * s;

        if (it != 2) {
            #pragma unroll
            for (int i = 0; i < 64; ++i) lg[i] = fmaf(prv[i], outd, lg[i]);
        }
    }

    // output layout (1, O, B, DO): contiguous per block
    out[((size_t)o * B_TOT + bt * BT + tb) * D_DIM + td] = outd;
}

// ---------------------------------------------------------------------------
extern "C" void kernel_launch(void* const* d_in, const int* in_sizes, int n_in,
                              void* d_out, int out_size, void* d_ws, size_t ws_size,
                              hipStream_t stream)
{
    (void)in_sizes; (void)n_in; (void)out_size; (void)ws_size;

    const float* x  = (const float*)d_in[0];   // (2048, 16, 64) f32
    const float* rw = (const float*)d_in[1];   // (32, 64, 16, 16) f32
    const float* nz = (const float*)d_in[2];   // (32, 64, 16, 16) f32
    float*       out = (float*)d_out;          // (1, 32, 2048, 16) f32

    // workspace: xp = 8 MB (A fragments), wf = 2 MB (B fragments)
    v2f* xp = (v2f*)d_ws;
    v2f* wfp = (v2f*)((char*)d_ws + (size_t)8 * 1024 * 1024);

    pack_x_kernel<<<4096, 256, 0, stream>>>(x, xp);
    pack_w_kernel<<<1024, 256, 0, stream>>>(rw, nz, wfp);

    dim3 grid(NBT, O_CAP);
    routing_fused_kernel<<<grid, 256, PR_BYTES, stream>>>(xp, wfp, out);
}